// SkipGramWord2Vec_47656957116685
// MI455X (gfx1250) — compile-verified
//
#include <hip/hip_runtime.h>

typedef __attribute__((ext_vector_type(2))) float v2f;
typedef __attribute__((ext_vector_type(8))) float v8f;

#define K_NEG  10
#define EMB    128
#define NACC   (K_NEG + 1)          // 11 accumulators (1 pos + 10 neg)
#define NDIAG  (NACC * 16)          // 176 diagonal values per tile

__device__ __forceinline__ float softplus(float x) {
    // -log_sigmoid(-x) = softplus(x), numerically stable
    float ax = __builtin_fabsf(x);
    return fmaxf(x, 0.0f) + log1pf(__expf(-ax));
}

// One wave32 handles a tile of 16 batch elements.
// WMMA f32 16x16x4: D = A(16 gathered rows) x B(center rows^T); diag(D) = needed dots.
__global__ __launch_bounds__(128) void sg_wmma_kernel(
    const int* __restrict__ center,
    const int* __restrict__ outside,
    const int* __restrict__ negids,
    const float* __restrict__ emb,
    float* __restrict__ partials)
{
    __shared__ float diag_lds[4][NDIAG];         // 4 waves/block * 176 floats

    const int lane = threadIdx.x & 31;
    const int wave = threadIdx.x >> 5;
    const int tile = blockIdx.x * 4 + wave;      // 16 batch elements per tile
    const int e    = lane & 15;                  // element within tile (row/col of matrices)
    const int half = lane >> 4;                  // K sub-pair within 4-chunk
    const int b    = tile * 16 + e;

    // Gathered row base pointers (per lane, duplicated across the two lane halves)
    const float* crow = emb + (size_t)center[b]  * EMB;
    const float* orow = emb + (size_t)outside[b] * EMB;
    const float* nrow[K_NEG];
#pragma unroll
    for (int j = 0; j < K_NEG; ++j)
        nrow[j] = emb + (size_t)negids[b * K_NEG + j] * EMB;

    const int coff = half * 2;   // A layout: lanes 0-15 -> K=0,1 ; lanes 16-31 -> K=2,3

    v8f pacc = {};
    v8f nacc[K_NEG];
#pragma unroll
    for (int j = 0; j < K_NEG; ++j) nacc[j] = (v8f){};

    // March over the embedding dimension in K=4 chunks (32 iterations).
    // 12 independent 8B/lane gathers + 11 WMMAs per chunk; compiler pipelines.
    for (int c = 0; c < EMB / 4; ++c) {
        const int col = c * 4 + coff;
        v2f bm = *(const v2f*)(crow + col);   // B matrix: center rows, transposed layout
        v2f ap = *(const v2f*)(orow + col);   // A matrix: outside rows
        pacc = __builtin_amdgcn_wmma_f32_16x16x4_f32(
            false, ap, false, bm, (short)0, pacc, false, false);
#pragma unroll
        for (int j = 0; j < K_NEG; ++j) {
            v2f an = *(const v2f*)(nrow[j] + col);
            nacc[j] = __builtin_amdgcn_wmma_f32_16x16x4_f32(
                false, an, false, bm, (short)0, nacc[j], false, false);
        }
    }

    // --- Diagonal extraction via LDS -------------------------------------
    // C/D layout: VGPR r, lanes 0-15 hold (M=r, N=lane); lanes 16-31 hold
    // (M=r+8, N=lane-16).  Element (m,m) is owned by lane m (m<8, vgpr m)
    // or lane m+16 (m>=8, vgpr m-8).  Owner condition: (lane&15) == r+8*half.
    float* my = diag_lds[wave];
#pragma unroll
    for (int r = 0; r < 8; ++r) {
        const int m = r + 8 * half;
        if ((lane & 15) == m) {
            my[0 * 16 + m] = -pacc[r];        // pos: loss = softplus(-pos_dot)
#pragma unroll
            for (int j = 0; j < K_NEG; ++j)
                my[(j + 1) * 16 + m] = nacc[j][r];   // neg: loss = softplus(+dot)
        }
    }
    // Same-wave LDS ops are in-order; make the ordering explicit and stop the
    // compiler from sinking the reads above the writes (CDNA5 split counter).
    asm volatile("s_wait_dscnt 0" ::: "memory");

    // Distribute the 176 softplus evals across the 32 lanes (~6 each).
    float lsum = 0.0f;
    for (int i = lane; i < NDIAG; i += 32)
        lsum += softplus(my[i]);

    // Deterministic butterfly reduction within the wave.
#pragma unroll
    for (int off = 16; off > 0; off >>= 1)
        lsum += __shfl_xor(lsum, off, 32);

    if (lane == 0) partials[tile] = lsum;
}

// Deterministic fixed-tree reduction of per-tile partials -> mean
__global__ __launch_bounds__(256) void sg_reduce_kernel(
    const float* __restrict__ partials, int n, float* __restrict__ out, float invB)
{
    __shared__ float s[256];
    float acc = 0.0f;
    for (int i = threadIdx.x; i < n; i += 256) acc += partials[i];
    s[threadIdx.x] = acc;
    __syncthreads();
    for (int off = 128; off > 0; off >>= 1) {
        if (threadIdx.x < off) s[threadIdx.x] += s[threadIdx.x + off];
        __syncthreads();
    }
    if (threadIdx.x == 0) out[0] = s[0] * invB;
}

extern "C" void kernel_launch(void* const* d_in, const int* in_sizes, int n_in,
                              void* d_out, int out_size, void* d_ws, size_t ws_size,
                              hipStream_t stream) {
    const int*   center  = (const int*)d_in[0];
    const int*   outside = (const int*)d_in[1];
    const int*   negids  = (const int*)d_in[2];
    const float* emb     = (const float*)d_in[3];
    float* out      = (float*)d_out;
    float* partials = (float*)d_ws;          // ntiles floats of scratch

    const int Bsz    = in_sizes[0];          // 65536
    const int ntiles = Bsz / 16;             // 4096 tiles (one wave each)

    sg_wmma_kernel<<<ntiles / 4, 128, 0, stream>>>(center, outside, negids, emb, partials);
    sg_reduce_kernel<<<1, 256, 0, stream>>>(partials, ntiles, out, 1.0f / (float)Bsz);
}